// diag_rtrl_61151744360733
// MI455X (gfx1250) — compile-verified
//
#include <hip/hip_runtime.h>
#include <cmath>

typedef __attribute__((ext_vector_type(2))) float v2f;
typedef __attribute__((ext_vector_type(4))) float v4f;
typedef __attribute__((ext_vector_type(8))) float v8f;

#define BETA 0.9f
#define BATCH 32
#define DIN   1024
#define DH    1024

// -----------------------------------------------------------------------------
// Kernel 1: u_new = BETA*u + x@W ; s = tanh(u_new)
// One 16x16 output tile per 128-thread block (4 wave32s). K split across the
// 4 waves (256 K-values each, 64 WMMA_F32_16X16X4_F32 steps), partials reduced
// via LDS, wave 0 does the fused epilogue.
//
// f32 WMMA operand layout (wave32, per ISA 7.12.2):
//   A 16x4:  lane L holds M=L%16; VGPR0/1 = K=(0,1) for lanes 0-15,
//            K=(2,3) for lanes 16-31.
//   B 4x16:  lane L holds N=L%16; VGPR0/1 = K=(0,1) for lanes 0-15,
//            K=(2,3) for lanes 16-31.
//   C/D 16x16: VGPR v holds row M = v + 8*(L/16), col N = L%16.
// -----------------------------------------------------------------------------
__global__ __launch_bounds__(128) void rtrl_gemm_kernel(
    const float* __restrict__ x, const float* __restrict__ W,
    const float* __restrict__ u, float* __restrict__ s_out,
    float* __restrict__ u_out)
{
  __shared__ float red[3][256];  // partial C tiles from waves 1..3

  const int lane   = threadIdx.x & 31;
  const int wave   = threadIdx.x >> 5;
  const int tile_n = blockIdx.x;   // 0..63  (DH/16)
  const int tile_m = blockIdx.y;   // 0..1   (BATCH/16)

  const int half = lane >> 4;      // 0: lanes 0-15, 1: lanes 16-31
  const int l16  = lane & 15;

  const int row = tile_m * 16 + l16;   // A-matrix row for this lane
  const int col = tile_n * 16 + l16;   // B-matrix col for this lane

  v8f acc = {};

  const int k0 = wave * (DIN / 4);
  const int k1 = k0 + (DIN / 4);

#pragma unroll 4
  for (int k = k0; k < k1; k += 4) {
    const int kk = k + 2 * half;
    // A: x[row, kk:kk+2] -- contiguous pair
    v2f a = *(const v2f*)(x + (size_t)row * DIN + kk);
    // B: W[kk, col], W[kk+1, col] -- column strided
    v2f b;
    b.x = W[(size_t)kk * DH + col];
    b.y = W[(size_t)(kk + 1) * DH + col];
    // 8 args: (neg_a, A, neg_b, B, c_mod, C, reuse_a, reuse_b)
    acc = __builtin_amdgcn_wmma_f32_16x16x4_f32(
        false, a, false, b, (short)0, acc, false, false);
  }

  // Cross-wave K reduction through LDS.
  if (wave > 0) {
#pragma unroll
    for (int v = 0; v < 8; ++v) red[wave - 1][v * 32 + lane] = acc[v];
  }
  __syncthreads();

  if (wave == 0) {
#pragma unroll
    for (int v = 0; v < 8; ++v) {
      float p = acc[v] + red[0][v * 32 + lane] + red[1][v * 32 + lane]
                       + red[2][v * 32 + lane];
      const int r = tile_m * 16 + v + 8 * half;   // C/D row for this VGPR
      const int c = tile_n * 16 + l16;
      const size_t idx = (size_t)r * DH + c;
      const float un = fmaf(BETA, u[idx], p);
      u_out[idx] = un;
      s_out[idx] = tanhf(un);
    }
  }
}

// -----------------------------------------------------------------------------
// Kernel 2: E_new[b,i,:] = BETA*E[b,i,:] + x[b,i]
// Bandwidth-bound stream (~268 MB -> ~11.5 us at 23.3 TB/s). One 256-thread
// block per (b,i) row; float4 per lane -> fully coalesced b128 traffic.
// E is single-use: non-temporal load/store (TH=NT) so 268 MB of streaming
// traffic does not evict the resident working set from the 192 MB L2.
// -----------------------------------------------------------------------------
__global__ __launch_bounds__(256) void rtrl_trace_kernel(
    const float* __restrict__ x, const float* __restrict__ E,
    float* __restrict__ E_out)
{
  const size_t rowIdx = blockIdx.x;             // b*DIN + i
  const float  xv     = x[rowIdx];              // scalar (uniform per block)
  const size_t base   = rowIdx * DH + (size_t)threadIdx.x * 4;

  v4f e = __builtin_nontemporal_load((const v4f*)(E + base));
  v4f r;
  r.x = fmaf(BETA, e.x, xv);
  r.y = fmaf(BETA, e.y, xv);
  r.z = fmaf(BETA, e.z, xv);
  r.w = fmaf(BETA, e.w, xv);
  __builtin_nontemporal_store(r, (v4f*)(E_out + base));
}

extern "C" void kernel_launch(void* const* d_in, const int* in_sizes, int n_in,
                              void* d_out, int out_size, void* d_ws, size_t ws_size,
                              hipStream_t stream) {
  (void)in_sizes; (void)n_in; (void)out_size; (void)d_ws; (void)ws_size;

  const float* x = (const float*)d_in[0];   // [32, 1024]
  const float* W = (const float*)d_in[1];   // [1024, 1024]
  const float* u = (const float*)d_in[2];   // [32, 1024]
  const float* E = (const float*)d_in[3];   // [32, 1024, 1024]

  float* out   = (float*)d_out;
  float* s_out = out;                                   // [32, 1024]
  float* u_out = out + (size_t)BATCH * DH;              // [32, 1024]
  float* E_out = out + (size_t)2 * BATCH * DH;          // [32, 1024, 1024]

  dim3 gemm_grid(DH / 16, BATCH / 16);                  // 64 x 2 tiles
  rtrl_gemm_kernel<<<gemm_grid, 128, 0, stream>>>(x, W, u, s_out, u_out);

  rtrl_trace_kernel<<<BATCH * DIN, 256, 0, stream>>>(x, E, E_out);
}